// SemanticRetriever_23948737642980
// MI455X (gfx1250) — compile-verified
//
#include <hip/hip_runtime.h>

#define D_DIM 128
#define BQ    32
#define KTOP  10
#define TILE_N 16
#define WAVES_PER_BLOCK 8
#define THREADS_P1 (WAVES_PER_BLOCK * 32)
#define MAX_WAVES 16384
#define NEG_INF (-__builtin_inff())

typedef float v2f __attribute__((ext_vector_type(2)));
typedef float v8f __attribute__((ext_vector_type(8)));

// ---------------------------------------------------------------------------
// Kernel 0: normalize the 32 queries into workspace (tiny).
// ---------------------------------------------------------------------------
__global__ void sr_normalize_q(const float* __restrict__ q, float* __restrict__ qn) {
    const int tid = threadIdx.x;                // 128 threads
    __shared__ float red[128];
    for (int b = 0; b < BQ; ++b) {
        float v = q[b * D_DIM + tid];
        red[tid] = v * v;
        __syncthreads();
        for (int s = 64; s > 0; s >>= 1) {
            if (tid < s) red[tid] += red[tid + s];
            __syncthreads();
        }
        float rn = rsqrtf(red[0] + 1e-12f);
        __syncthreads();
        qn[b * D_DIM + tid] = v * rn;
    }
}

// ---------------------------------------------------------------------------
// Kernel 1: WMMA scoring + per-wave private top-K.
// Each wave owns a strided set of 16-row corpus tiles. Per tile:
//   - register-stage the whole per-lane corpus row (32 x b64 loads in flight),
//   - 32 steps of V_WMMA_F32_16X16X4_F32 (x2 for 32 queries), row sum-of-
//     squares folded into the staged data (no extra memory traffic),
//   - scale by rsqrt(row sumsq), stage via wave-local LDS, lane L updates a
//     register-resident sorted top-10 for query L.
// ---------------------------------------------------------------------------
__global__ void sr_score_topk_pass1(const float* __restrict__ corpus,
                                    const float* __restrict__ qn,
                                    float* __restrict__ candVal,
                                    int*   __restrict__ candIdx,
                                    int numTiles, int totalWaves, int N) {
    __shared__ float sQ[BQ * D_DIM];                         // 16 KB
    __shared__ float sScore[WAVES_PER_BLOCK][BQ][TILE_N];    // 16 KB

    const int tid = threadIdx.x;
    for (int i = tid; i < BQ * D_DIM; i += THREADS_P1) sQ[i] = qn[i];
    __syncthreads();

    const int wave = tid >> 5;
    const int lane = tid & 31;
    const int half = lane >> 4;    // 0: K = k0,k0+1   1: K = k0+2,k0+3
    const int sub  = lane & 15;    // A: query row 0..15 ; B: corpus col 0..15
    const int gw   = blockIdx.x * WAVES_PER_BLOCK + wave;

    float vals[KTOP];
    int   idxs[KTOP];
#pragma unroll
    for (int i = 0; i < KTOP; ++i) { vals[i] = NEG_INF; idxs[i] = 0; }

    float (*sc)[TILE_N] = sScore[wave];

    for (int t = gw; t < numTiles; t += totalWaves) {
        const int rowBase = t * TILE_N;
        int row = rowBase + sub;
        if (row >= N) row = N - 1;                 // clamp (N % 16 == 0 normally)
        const float* crow = corpus + (size_t)row * D_DIM;

        // Prefetch next tile (2 lines/lane cover all 4 cachelines of a row).
        {
            const int tn = t + totalWaves;
            if (tn < numTiles) {
                int rown = tn * TILE_N + sub;
                if (rown >= N) rown = N - 1;
                const char* p = (const char*)(corpus + (size_t)rown * D_DIM) + half * 256;
                __builtin_prefetch(p, 0, 1);
                __builtin_prefetch(p + 128, 0, 1);
            }
        }

        // Register-stage the per-lane corpus data: 32 b64 loads in flight.
        v2f breg[32];
#pragma unroll
        for (int k0 = 0; k0 < D_DIM; k0 += 4)
            breg[k0 >> 2] = *(const v2f*)&crow[k0 + (half << 1)];

        v8f acc0 = {};
        v8f acc1 = {};
        float ss = 0.f;
#pragma unroll
        for (int k0 = 0; k0 < D_DIM; k0 += 4) {
            const int kk = k0 + (half << 1);
            v2f a0 = *(const v2f*)&sQ[sub * D_DIM + kk];          // queries 0..15
            v2f a1 = *(const v2f*)&sQ[(sub + 16) * D_DIM + kk];   // queries 16..31
            v2f b  = breg[k0 >> 2];                               // corpus (B operand)
            ss += b.x * b.x + b.y * b.y;                          // free row-norm partial
            acc0 = __builtin_amdgcn_wmma_f32_16x16x4_f32(false, a0, false, b,
                                                         (short)0, acc0, false, false);
            acc1 = __builtin_amdgcn_wmma_f32_16x16x4_f32(false, a1, false, b,
                                                         (short)0, acc1, false, false);
        }
        ss += __shfl_xor(ss, 16, 32);              // combine half-wave K partitions
        const float rn = rsqrtf(ss + 1e-12f);

        // D-layout: acc0[v] = S[q = v + 8*half][n = sub]; column n scale = rn.
#pragma unroll
        for (int v = 0; v < 8; ++v) {
            sc[v + 8 * half][sub]      = acc0[v] * rn;
            sc[16 + v + 8 * half][sub] = acc1[v] * rn;
        }
        asm volatile("s_wait_dscnt 0x0" ::: "memory");   // wave-local LDS visibility

        // lane L owns query L: scan the 16 candidates of this tile.
        const float* myrow = sc[lane];
#pragma unroll
        for (int j = 0; j < TILE_N; ++j) {
            float v = myrow[j];
            int   n = rowBase + j;
            if (n < N && v > vals[KTOP - 1]) {
                vals[KTOP - 1] = v;
                idxs[KTOP - 1] = n;
#pragma unroll
                for (int i2 = KTOP - 1; i2 > 0; --i2) {
                    if (vals[i2] > vals[i2 - 1]) {
                        float tv = vals[i2]; vals[i2] = vals[i2 - 1]; vals[i2 - 1] = tv;
                        int   ti = idxs[i2]; idxs[i2] = idxs[i2 - 1]; idxs[i2 - 1] = ti;
                    }
                }
            }
        }
        asm volatile("s_wait_dscnt 0x0" ::: "memory");   // reads done before next tile writes
    }

    // Emit this wave's per-query top-10 candidates (always fully written).
    const size_t base = (size_t)gw * (BQ * KTOP) + (size_t)lane * KTOP;
#pragma unroll
    for (int i = 0; i < KTOP; ++i) {
        candVal[base + i] = vals[i];
        candIdx[base + i] = idxs[i];
    }
}

// ---------------------------------------------------------------------------
// Kernel 2: one block per query. Merge W*10 candidates -> top-10, write
// values, indices (as float), and gathered normalized corpus rows.
// ---------------------------------------------------------------------------
__global__ void sr_final_topk(const float* __restrict__ corpus,
                              const float* __restrict__ candVal,
                              const int*   __restrict__ candIdx,
                              float* __restrict__ out,
                              int totalWaves) {
    const int q   = blockIdx.x;     // 0..31
    const int tid = threadIdx.x;    // 256
    __shared__ float lv[256 * KTOP];
    __shared__ int   li[256 * KTOP];
    __shared__ float rv[256];
    __shared__ int   ri[256];
    __shared__ float fv[KTOP];
    __shared__ int   fi[KTOP];

    float vals[KTOP];
    int   idxs[KTOP];
#pragma unroll
    for (int i = 0; i < KTOP; ++i) { vals[i] = NEG_INF; idxs[i] = 0; }

    const int E = totalWaves * KTOP;
    for (int e = tid; e < E; e += 256) {
        const int gw = e / KTOP;
        const int i  = e - gw * KTOP;
        const size_t off = (size_t)gw * (BQ * KTOP) + (size_t)q * KTOP + i;
        float v = candVal[off];
        if (v > vals[KTOP - 1]) {
            int n = candIdx[off];
            vals[KTOP - 1] = v;
            idxs[KTOP - 1] = n;
#pragma unroll
            for (int i2 = KTOP - 1; i2 > 0; --i2) {
                if (vals[i2] > vals[i2 - 1]) {
                    float tv = vals[i2]; vals[i2] = vals[i2 - 1]; vals[i2 - 1] = tv;
                    int   ti = idxs[i2]; idxs[i2] = idxs[i2 - 1]; idxs[i2 - 1] = ti;
                }
            }
        }
    }
#pragma unroll
    for (int i = 0; i < KTOP; ++i) {
        lv[tid * KTOP + i] = vals[i];
        li[tid * KTOP + i] = idxs[i];
    }
    __syncthreads();

    // 10 rounds of block-wide argmax over the 2560 staged candidates.
    for (int r = 0; r < KTOP; ++r) {
        float bv = NEG_INF;
        int   bs = tid * KTOP;
#pragma unroll
        for (int j = 0; j < KTOP; ++j) {
            float v = lv[tid * KTOP + j];
            if (v > bv) { bv = v; bs = tid * KTOP + j; }
        }
        rv[tid] = bv; ri[tid] = bs;
        __syncthreads();
        for (int s = 128; s > 0; s >>= 1) {
            if (tid < s && rv[tid + s] > rv[tid]) { rv[tid] = rv[tid + s]; ri[tid] = ri[tid + s]; }
            __syncthreads();
        }
        if (tid == 0) {
            fv[r] = rv[0];
            fi[r] = li[ri[0]];
            lv[ri[0]] = NEG_INF;
        }
        __syncthreads();
    }

    // top_vals [32,10] then top_idx [32,10] (as float).
    if (tid < KTOP) {
        out[q * KTOP + tid]             = fv[tid];
        out[BQ * KTOP + q * KTOP + tid] = (float)fi[tid];
    }
    __syncthreads();

    // gathered [32,10,128]: reload winner rows, renormalize, write.
    for (int i = 0; i < KTOP; ++i) {
        const int idx = fi[i];
        float v = 0.f;
        if (tid < D_DIM) v = corpus[(size_t)idx * D_DIM + tid];
        rv[tid] = v * v;
        __syncthreads();
        for (int s = 128; s > 0; s >>= 1) {
            if (tid < s) rv[tid] += rv[tid + s];
            __syncthreads();
        }
        const float rn = rsqrtf(rv[0] + 1e-12f);
        __syncthreads();
        if (tid < D_DIM)
            out[(size_t)BQ * KTOP * 2 + ((size_t)q * KTOP + i) * D_DIM + tid] = v * rn;
        __syncthreads();
    }
}

// ---------------------------------------------------------------------------
extern "C" void kernel_launch(void* const* d_in, const int* in_sizes, int n_in,
                              void* d_out, int out_size, void* d_ws, size_t ws_size,
                              hipStream_t stream) {
    const float* query  = (const float*)d_in[0];
    const float* corpus = (const float*)d_in[1];
    const int N = in_sizes[1] / D_DIM;
    float* out = (float*)d_out;

    // Workspace layout: [ qn 32x128 f32 | candVal W*320 f32 | candIdx W*320 i32 ]
    char*  ws      = (char*)d_ws;
    float* qn      = (float*)ws;
    const size_t qnBytes = (size_t)BQ * D_DIM * sizeof(float);
    const size_t perWave = (size_t)BQ * KTOP * (sizeof(float) + sizeof(int)); // 2560 B
    size_t avail = (ws_size > qnBytes + 64) ? (ws_size - qnBytes - 64) : 0;
    long wmax = (long)(avail / perWave);
    int totalWaves = (int)(wmax < 1 ? 1 : (wmax > MAX_WAVES ? MAX_WAVES : wmax));
    int NB = totalWaves / WAVES_PER_BLOCK;
    if (NB < 1) NB = 1;
    totalWaves = NB * WAVES_PER_BLOCK;

    float* candVal = (float*)(ws + qnBytes);
    int*   candIdx = (int*)(ws + qnBytes + (size_t)totalWaves * BQ * KTOP * sizeof(float));
    const int numTiles = (N + TILE_N - 1) / TILE_N;

    sr_normalize_q<<<1, 128, 0, stream>>>(query, qn);
    sr_score_topk_pass1<<<NB, THREADS_P1, 0, stream>>>(corpus, qn, candVal, candIdx,
                                                       numTiles, totalWaves, N);
    sr_final_topk<<<BQ, 256, 0, stream>>>(corpus, candVal, candIdx, out, totalWaves);
}